// RegressionGradCamModel_69690139344957
// MI455X (gfx1250) — compile-verified
//
#include <hip/hip_runtime.h>
#include <hip/hip_bf16.h>

typedef __attribute__((ext_vector_type(2))) float v2f;
typedef __attribute__((ext_vector_type(8))) float v8f;

#define BATCH 32
#define HIMG  512
#define WIMG  512
#define CIN   3
#define FCH   16
#define POOL  32
#define HP    16   // pooled h
#define WP    16   // pooled w

// ---------------------------------------------------------------------------
// Phase 1: fused avg-pool + (W1·w_head) channel dot.
// cam_small[b,i,j] = (1/(256*1024)) * sum_{32x32 px, c} inputs * v[c]
// One block per pooled cell: 8192 blocks x 256 threads, float4-coalesced.
// Memory: 100.7 MB streamed once -> the dominant cost (~4.3 us at 23.3 TB/s).
// ---------------------------------------------------------------------------
__global__ void pool_dot_kernel(const float* __restrict__ in,
                                const float* __restrict__ W1,
                                const float* __restrict__ w_head,
                                float* __restrict__ cam_small)
{
    const int blk = blockIdx.x;          // 0..8191 = b*256 + i*16 + j
    const int j = blk & 15;
    const int i = (blk >> 4) & 15;
    const int b = blk >> 8;

    // v[c] = sum_f W1[c,f] * w_head[f]   (uniform -> scalar loads, broadcast)
    float v0 = 0.f, v1 = 0.f, v2 = 0.f;
#pragma unroll
    for (int f = 0; f < FCH; ++f) {
        const float wh = w_head[f];
        v0 += W1[0 * FCH + f] * wh;
        v1 += W1[1 * FCH + f] * wh;
        v2 += W1[2 * FCH + f] * wh;
    }
    const float vc[3] = {v0, v1, v2};

    // Tile: rows i*32..+31, cols j*32..+31, 3 ch -> 96 floats/row = 24 float4.
    const size_t rowStride = (size_t)WIMG * CIN;  // 1536 floats
    const size_t base = ((size_t)b * HIMG + (size_t)i * POOL) * rowStride
                      + (size_t)j * POOL * CIN;

    float acc = 0.f;
#pragma unroll
    for (int k = 0; k < 3; ++k) {
        const int idx = threadIdx.x + k * 256;   // 0..767 over (row, float4)
        const int row = idx / 24;
        const int c4  = idx % 24;
        const float4 x = *(const float4*)(in + base + (size_t)row * rowStride + c4 * 4);
        const int ch = c4 % 3;                   // channel of element 0 ((4*c4)%3 == c4%3)
        acc += x.x * vc[ch];
        acc += x.y * vc[(ch + 1) % 3];
        acc += x.z * vc[(ch + 2) % 3];
        acc += x.w * vc[ch];
    }

    __shared__ float red[256];
    red[threadIdx.x] = acc;
    __syncthreads();
#pragma unroll
    for (int s = 128; s > 0; s >>= 1) {
        if ((int)threadIdx.x < s) red[threadIdx.x] += red[threadIdx.x + s];
        __syncthreads();
    }
    if (threadIdx.x == 0) cam_small[blk] = red[0] * (1.0f / 262144.0f);
}

// ---------------------------------------------------------------------------
// Bilinear interpolation weights, half-pixel convention, edge clamp.
// scale = 16/512 = 1/32.  U[dst, k] = uweight(k, i0, i1, w).
// ---------------------------------------------------------------------------
__device__ __forceinline__ void interp16(int dst, int& i0, int& i1, float& w)
{
    const float s  = ((float)dst + 0.5f) * 0.03125f - 0.5f;
    const float fs = floorf(s);
    w  = s - fs;
    i0 = (int)fs;
    i1 = i0 + 1;
    i0 = i0 < 0  ? 0  : i0;
    i1 = i1 > 15 ? 15 : i1;   // clamp; i0==i1 at edges => total weight 1
}

__device__ __forceinline__ float uweight(int k, int i0, int i1, float w)
{
    float r = 0.f;
    if (k == i0) r += 1.0f - w;
    if (k == i1) r += w;
    return r;
}

// ---------------------------------------------------------------------------
// Phase 2: T[b, Y, j] = sum_i U[Y,i] * cam_small[b,i,j]   (U generated in-reg)
// One wave per (b, Ytile): 1024 waves, each 4x V_WMMA_F32_16X16X4_F32 (K=16).
// A = U rows (16x4 chunk), B = cam_small[b] (4x16 chunk), D += A*B.
// ---------------------------------------------------------------------------
__global__ void rows_wmma_kernel(const float* __restrict__ cam_small,
                                 float* __restrict__ T)
{
    const int blk  = blockIdx.x;        // b*32 + t
    const int t    = blk & 31;          // Y tile
    const int b    = blk >> 5;
    const int lane = threadIdx.x;       // blockDim.x == 32, EXEC all ones
    const int m    = lane & 15;         // A: M row / B: N col
    const int kb   = (lane >> 4) << 1;  // K base for this half-wave (0 or 2)

    const int Y = t * 16 + m;
    int i0, i1; float w;
    interp16(Y, i0, i1, w);

    const float* cs = cam_small + (size_t)b * 256;   // 16x16, row = i, col = j
    v8f acc = {};
#pragma unroll
    for (int kc = 0; kc < 4; ++kc) {
        const int k0 = kc * 4 + kb;
        v2f a, bm;
        a.x  = uweight(k0,     i0, i1, w);
        a.y  = uweight(k0 + 1, i0, i1, w);
        bm.x = cs[k0 * 16 + m];          // B[K=k0][N=m]
        bm.y = cs[(k0 + 1) * 16 + m];
        acc = __builtin_amdgcn_wmma_f32_16x16x4_f32(
                  false, a, false, bm, (short)0, acc, false, false);
    }

    // D layout: VGPR v -> M = v + 8*(lane>=16), N = m
    float* Tb = T + ((size_t)b * 512 + (size_t)t * 16) * 16;
    const int mhi = (lane >> 4) * 8;
#pragma unroll
    for (int v = 0; v < 8; ++v)
        Tb[(v + mhi) * 16 + m] = acc[v];
}

// ---------------------------------------------------------------------------
// Phase 3: cam[b, Y, X] = sum_j T[b,Y,j] * U[X,j]  (the transposed resize).
// 8 waves/block, one 16x16 output tile per wave: 32768 tiles, 131072 WMMAs.
// Output-store bound: 33.5 MB -> ~1.4 us at 23.3 TB/s.
// ---------------------------------------------------------------------------
__global__ void cols_wmma_kernel(const float* __restrict__ T,
                                 float* __restrict__ out)
{
    const int blk  = blockIdx.x;        // b*128 + t*4 + xg
    const int xg   = blk & 3;
    const int t    = (blk >> 2) & 31;   // Y tile
    const int b    = blk >> 7;
    const int wave = threadIdx.x >> 5;
    const int lane = threadIdx.x & 31;
    const int xt   = xg * 8 + wave;     // X tile 0..31
    const int m    = lane & 15;
    const int kb   = (lane >> 4) << 1;

    const int X = xt * 16 + m;          // B: N = m -> output column X
    int j0, j1; float w;
    interp16(X, j0, j1, w);

    const float* Trow = T + ((size_t)b * 512 + (size_t)t * 16) * 16;  // 16x16 A tile
    v8f acc = {};
#pragma unroll
    for (int kc = 0; kc < 4; ++kc) {
        const int k0 = kc * 4 + kb;
        v2f a, bm;
        a.x  = Trow[m * 16 + k0];        // A[M=m][K=k0]
        a.y  = Trow[m * 16 + k0 + 1];
        bm.x = uweight(k0,     j0, j1, w);
        bm.y = uweight(k0 + 1, j0, j1, w);
        acc = __builtin_amdgcn_wmma_f32_16x16x4_f32(
                  false, a, false, bm, (short)0, acc, false, false);
    }

    float* ob = out + (size_t)b * HIMG * WIMG;
    const int mhi = (lane >> 4) * 8;
#pragma unroll
    for (int v = 0; v < 8; ++v) {
        const int Yr = t * 16 + v + mhi;
        ob[(size_t)Yr * WIMG + (size_t)xt * 16 + m] = acc[v];
    }
}

// ---------------------------------------------------------------------------
extern "C" void kernel_launch(void* const* d_in, const int* in_sizes, int n_in,
                              void* d_out, int out_size, void* d_ws, size_t ws_size,
                              hipStream_t stream)
{
    const float* in  = (const float*)d_in[0];   // (32,512,512,3)
    const float* W1  = (const float*)d_in[1];   // (3,16)
    const float* wh  = (const float*)d_in[2];   // (16,)
    float* out       = (float*)d_out;           // (32,512,512,1)

    float* cam_small = (float*)d_ws;            // 8192 floats  (32 KB)
    float* T         = cam_small + 8192;        // 262144 floats (1 MB)

    pool_dot_kernel <<<BATCH * HP * WP, 256, 0, stream>>>(in, W1, wh, cam_small);
    rows_wmma_kernel<<<BATCH * 32,       32, 0, stream>>>(cam_small, T);
    cols_wmma_kernel<<<BATCH * 32 * 4,  256, 0, stream>>>(T, out);
}